// MultiHeadedAttention_16260746182850
// MI455X (gfx1250) — compile-verified
//
#include <hip/hip_runtime.h>
#include <hip/hip_bf16.h>

typedef __attribute__((ext_vector_type(2))) float v2f;
typedef __attribute__((ext_vector_type(8))) float v8f;

#define B_SZ   8
#define S_LEN  2048
#define D_DIM  256
#define DK_DIM 128

// D = A(16x4) * B(4x16) + C(16x16), all fp32, full wave32 cooperative.
__device__ __forceinline__ v8f wmma_f32(v2f a, v2f b, v8f c) {
  return __builtin_amdgcn_wmma_f32_16x16x4_f32(false, a, false, b, (short)0, c,
                                               false, false);
}

// out[r_dst][c] = ropeOpt( src[r_src][:] @ W^T + bias ), one 16-row tile per block.
// 8 waves x (2 col-tiles of 16) = 256 output columns.
__global__ void __launch_bounds__(256) proj_rope_kernel(
    const float* __restrict__ src, const float* __restrict__ W,
    const float* __restrict__ bias, float* __restrict__ dst,
    int src_tile_stride, int dst_tile_stride, int do_rope) {
  __shared__ float xs[16][260];  // 260 pad: LDS bank-conflict-free A reads
  const int tid = threadIdx.x;
  const int tile_r = blockIdx.x;
  const size_t src_r0 = (size_t)tile_r * src_tile_stride;

  // Stage the 16x256 activation tile (coalesced b128 loads -> LDS).
  for (int i = tid; i < 16 * 64; i += 256) {
    const int row = i >> 6, c4 = (i & 63) << 2;
    *(float4*)&xs[row][c4] = *(const float4*)&src[(src_r0 + row) * D_DIM + c4];
  }
  __syncthreads();

  const int wave = tid >> 5, lane = tid & 31;
  const int ar  = lane & 15;          // M (or N) index within half-wave
  const int ko  = (lane >> 4) << 1;   // K sub-offset: lanes 0-15 -> K0,1 ; 16-31 -> K2,3
  const int mhi = (lane >> 4) << 3;   // C-matrix row offset (+8 for upper lanes)
  const int cbase = wave << 5;

  v8f acc0 = {}; v8f acc1 = {};
  for (int kk = 0; kk < D_DIM; kk += 4) {
    v2f a, b0, b1;
    { float2 t = *(const float2*)&xs[ar][kk + ko];                              a.x = t.x;  a.y = t.y;  }
    { float2 t = *(const float2*)&W[(size_t)(cbase + ar) * D_DIM + kk + ko];    b0.x = t.x; b0.y = t.y; }
    { float2 t = *(const float2*)&W[(size_t)(cbase + 16 + ar) * D_DIM + kk + ko]; b1.x = t.x; b1.y = t.y; }
    acc0 = wmma_f32(a, b0, acc0);
    acc1 = wmma_f32(a, b1, acc1);
  }

  const int m_base = (int)(src_r0 % S_LEN);  // sequence position of tile row 0
  const size_t dst_r0 = (size_t)tile_r * dst_tile_stride;
  for (int t = 0; t < 2; ++t) {
    v8f acc = t ? acc1 : acc0;
    const int col = cbase + t * 16 + ar;
    // theta = exp(-(2*ln(1e4)/DK) * floor(d/2)), d = col % DK
    const float th = __expf(-0.14391156831f * (float)((col & (DK_DIM - 1)) >> 1));
    for (int j = 0; j < 8; ++j) {
      const int M = j + mhi;
      float val = acc[j] + bias[col];
      if (do_rope) {
        const float pv = __shfl_xor(val, 1);  // RoPE partner lives in lane^1
        const float ang = (float)(m_base + M) * th;
        const float cs = __cosf(ang), sn = __sinf(ang);
        val = (col & 1) ? (val * cs + pv * sn) : (val * cs - pv * sn);
      }
      dst[(dst_r0 + M) * D_DIM + col] = val;
    }
  }
}

// One workgroup per (b,h). Scores 16x2048 live in dynamic LDS (131 KB < 320 KB WGP LDS).
__global__ void __launch_bounds__(256) attn_kernel(
    const float* __restrict__ q_ws, const float* __restrict__ k_ws,
    const float* __restrict__ v_ws, float* __restrict__ y_ws) {
  extern __shared__ float sc[];  // [16][2052] padded row stride
  const int b = blockIdx.x >> 1;
  const int h = blockIdx.x & 1;
  const int tid = threadIdx.x;
  const int wave = tid >> 5, lane = tid & 31;
  const int ar  = lane & 15;
  const int ko  = (lane >> 4) << 1;
  const int mhi = (lane >> 4) << 3;

  const float* qp = q_ws + (size_t)b * 16 * D_DIM + h * DK_DIM;
  const float* kp = k_ws + (size_t)b * S_LEN * D_DIM + h * DK_DIM;
  const float* vp = v_ws + (size_t)b * S_LEN * D_DIM + h * DK_DIM;

  // Phase 1: scores = (q @ k^T) * DK^-0.5 ; each wave does 16 kv-tiles.
  for (int t = wave; t < S_LEN / 16; t += 8) {
    v8f acc = {};
    const int kv0 = t << 4;
    for (int kk = 0; kk < DK_DIM; kk += 4) {
      v2f a, bb;
      { float2 x = *(const float2*)&qp[(size_t)ar * D_DIM + kk + ko];          a.x = x.x;  a.y = x.y;  }
      { float2 x = *(const float2*)&kp[(size_t)(kv0 + ar) * D_DIM + kk + ko];  bb.x = x.x; bb.y = x.y; }
      acc = wmma_f32(a, bb, acc);
    }
    for (int j = 0; j < 8; ++j)
      sc[(j + mhi) * 2052 + kv0 + ar] = acc[j] * 0.08838834764831845f;
  }
  __syncthreads();

  // Phase 2: per-row softmax; fold ablation (rows>=10 zero, drop row 3/7) and
  // per-head scale into the normalization coefficient.
  const float ab  = h ? 0.8f : 1.0f;
  const int   abl = h ? 7 : 3;
  for (int r = wave; r < 16; r += 8) {
    float* row = sc + r * 2052;
    float mx = -3.4e38f;
    for (int i = lane; i < S_LEN; i += 32) mx = fmaxf(mx, row[i]);
    for (int o = 16; o > 0; o >>= 1) mx = fmaxf(mx, __shfl_xor(mx, o));
    float sum = 0.f;
    for (int i = lane; i < S_LEN; i += 32) {
      float e = __expf(row[i] - mx); row[i] = e; sum += e;
    }
    for (int o = 16; o > 0; o >>= 1) sum += __shfl_xor(sum, o);
    const float cf = ((r < 10) && (r != abl)) ? (ab / sum) : 0.f;
    for (int i = lane; i < S_LEN; i += 32) row[i] *= cf;
  }
  __syncthreads();

  // Phase 3: y = O @ v ; each wave owns one 16-wide slice of DK.
  const int d0 = wave << 4;
  v8f acc = {};
  for (int kk = 0; kk < S_LEN; kk += 4) {
    v2f a, bb;
    { float2 x = *(const float2*)&sc[ar * 2052 + kk + ko]; a.x = x.x; a.y = x.y; }
    bb.x = vp[(size_t)(kk + ko) * D_DIM + d0 + ar];
    bb.y = vp[(size_t)(kk + ko + 1) * D_DIM + d0 + ar];
    acc = wmma_f32(a, bb, acc);
  }
  float* yp = y_ws + (size_t)b * 16 * D_DIM + h * DK_DIM + d0;
  for (int j = 0; j < 8; ++j)
    yp[(size_t)(j + mhi) * D_DIM + ar] = acc[j];
}

// out[b,s,:] = bo for all s >= 16 (those rows have y == 0 exactly).
__global__ void __launch_bounds__(256) fill_bias_kernel(
    const float* __restrict__ bias, float* __restrict__ out) {
  const size_t total = (size_t)B_SZ * (S_LEN - 16) * (D_DIM / 4);
  const size_t idx = (size_t)blockIdx.x * 256 + threadIdx.x;
  if (idx >= total) return;
  const int    c4  = (int)(idx & 63);
  const size_t row = idx >> 6;
  const size_t s   = 16 + (row % (S_LEN - 16));
  const size_t bb  = row / (S_LEN - 16);
  ((float4*)out)[(bb * S_LEN + s) * (D_DIM / 4) + c4] = ((const float4*)bias)[c4];
}

extern "C" void kernel_launch(void* const* d_in, const int* in_sizes, int n_in,
                              void* d_out, int out_size, void* d_ws, size_t ws_size,
                              hipStream_t stream) {
  const float* query  = (const float*)d_in[0];
  const float* keys   = (const float*)d_in[1];
  const float* values = (const float*)d_in[2];
  const float* Wq = (const float*)d_in[3];
  const float* bq = (const float*)d_in[4];
  const float* Wk = (const float*)d_in[5];
  const float* bk = (const float*)d_in[6];
  const float* Wv = (const float*)d_in[7];
  const float* bv = (const float*)d_in[8];
  const float* Wo = (const float*)d_in[9];
  const float* bo = (const float*)d_in[10];
  float* out = (float*)d_out;

  char* ws = (char*)d_ws;
  float* q_ws = (float*)(ws);                                   // 8*16*256 f32
  float* y_ws = (float*)(ws + (size_t)32768 * 4);               // 8*16*256 f32
  float* k_ws = (float*)(ws + (size_t)2 * 32768 * 4);           // 8*2048*256 f32
  float* v_ws = (float*)(ws + (size_t)2 * 32768 * 4 + (size_t)B_SZ * S_LEN * D_DIM * 4);

  // q: only rows 0..15 of each batch are ever used downstream.
  proj_rope_kernel<<<B_SZ, 256, 0, stream>>>(query, Wq, bq, q_ws, S_LEN, 16, 1);
  proj_rope_kernel<<<B_SZ * S_LEN / 16, 256, 0, stream>>>(keys,   Wk, bk, k_ws, 16, 16, 1);
  proj_rope_kernel<<<B_SZ * S_LEN / 16, 256, 0, stream>>>(values, Wv, bv, v_ws, 16, 16, 1);
  attn_kernel<<<B_SZ * 2, 256, 16 * 2052 * 4, stream>>>(q_ws, k_ws, v_ws, y_ws);
  // Output projection for rows 0..15 per batch (rows 10..15 of y are zero -> bo).
  proj_rope_kernel<<<B_SZ, 256, 0, stream>>>(y_ws, Wo, bo, out, 16, S_LEN, 0);
  fill_bias_kernel<<<4064, 256, 0, stream>>>(bo, out);
}